// LinkPredictor_32908039422280
// MI455X (gfx1250) — compile-verified
//
#include <hip/hip_runtime.h>

#define B_N 2048
#define D_N 512
#define R_N 32
#define TSTR 520  // padded LDS row stride in floats (keeps 8B alignment, rotates banks)

typedef float v2f __attribute__((ext_vector_type(2)));
typedef float v8f __attribute__((ext_vector_type(8)));

__global__ void zero_counts_kernel(int* __restrict__ counts) {
    if (threadIdx.x < R_N) counts[threadIdx.x] = 0;
}

__global__ void scatter_kernel(const int* __restrict__ rel_id,
                               int* __restrict__ counts,
                               int* __restrict__ idx) {
    int b = blockIdx.x * blockDim.x + threadIdx.x;
    if (b < B_N) {
        int r = rel_id[b];
        int pos = atomicAdd(&counts[r], 1);
        idx[r * B_N + pos] = b;
    }
}

__launch_bounds__(256)
__global__ void score_kernel(const float* __restrict__ head,
                             const float* __restrict__ tail,
                             const float* __restrict__ W,
                             const float* __restrict__ bias,
                             const int* __restrict__ counts,
                             const int* __restrict__ idx,
                             float* __restrict__ out) {
    const int r    = blockIdx.x;
    const int tile = blockIdx.y;
    const int cnt  = counts[r];
    if (tile * 16 >= cnt) return;   // uniform per block

    __shared__ float sT[16 * TSTR];   // gathered tail tile [16][512] (padded)
    __shared__ int   sIdx[16];        // real sample index (-1 for padding)
    __shared__ int   sIdxSafe[16];    // clamped index (0 for padding) -> branchless loads
    __shared__ float sPart[8][16];    // per-wave partial scores for 16 samples
    __shared__ float sBias;

    const int tid = threadIdx.x;

    if (tid < 16) {
        int j = tile * 16 + tid;
        int s = (j < cnt) ? idx[r * B_N + j] : -1;
        sIdx[tid] = s;
        sIdxSafe[tid] = (s >= 0) ? s : 0;
    }
    __syncthreads();

    // Stage tail tile into LDS. Padding rows are zero-filled, so their WMMA
    // output rows are exactly 0.0 -> head gather below needs no mask.
    for (int i = tid; i < 16 * D_N; i += 256) {
        int row = i >> 9;          // / 512
        int col = i & (D_N - 1);
        int s = sIdx[row];
        sT[row * TSTR + col] = (s >= 0) ? tail[(size_t)s * D_N + col] : 0.0f;
    }

    // Deterministic bias sum: wave 0 strided-sums then xor-tree reduces
    if (tid < 32) {
        float p = 0.0f;
        for (int e = tid; e < D_N; e += 32) p += bias[(size_t)r * D_N + e];
        for (int off = 16; off >= 1; off >>= 1) p += __shfl_xor(p, off, 32);
        if (tid == 0) sBias = p;
    }
    __syncthreads();

    const int wave = tid >> 5;      // 8 waves
    const int lane = tid & 31;
    const int half = lane >> 4;     // 0: lanes 0-15, 1: lanes 16-31
    const int n16  = lane & 15;     // M index for A, N index for B/C
    const int koff = half * 2;      // K sub-offset within 4-wide step

    const float* tPtr = &sT[n16 * TSTR + koff];

    // Preload this lane's 8 (clamped) sample row bases for the head gather
    const float* hRow[8];
#pragma unroll
    for (int v = 0; v < 8; ++v)
        hRow[v] = head + (size_t)sIdxSafe[v + 8 * half] * D_N;

    float acc[8];
#pragma unroll
    for (int v = 0; v < 8; ++v) acc[v] = 0.0f;

#pragma unroll
    for (int p = 0; p < 2; ++p) {
        const int eb0 = (wave * 4 + p * 2) * 16;   // this wave's e-block pair
        const int eb1 = eb0 + 16;

        // B operand: W row e = eb + n16, contiguous float2 at d = k + koff
        const v2f* wp0 = reinterpret_cast<const v2f*>(
            W + ((size_t)r * D_N + (eb0 + n16)) * D_N + koff);
        const v2f* wp1 = reinterpret_cast<const v2f*>(
            W + ((size_t)r * D_N + (eb1 + n16)) * D_N + koff);

        v8f c0 = {0.f, 0.f, 0.f, 0.f, 0.f, 0.f, 0.f, 0.f};
        v8f c1 = {0.f, 0.f, 0.f, 0.f, 0.f, 0.f, 0.f, 0.f};

#pragma unroll 8
        for (int k = 0; k < D_N; k += 4) {
            v2f a  = *reinterpret_cast<const v2f*>(tPtr + k);
            v2f b0 = wp0[k >> 1];
            v2f b1 = wp1[k >> 1];
            // D = A(16x4 f32) * B(4x16 f32) + C  -> v_wmma_f32_16x16x4_f32
            c0 = __builtin_amdgcn_wmma_f32_16x16x4_f32(
                     false, a, false, b0, (short)0, c0, false, false);
            c1 = __builtin_amdgcn_wmma_f32_16x16x4_f32(
                     false, a, false, b1, (short)0, c1, false, false);
        }

        // Fuse head dot-product: C layout lane=(n), vgpr v -> row m = v + 8*half.
        // Branchless: pad rows have c==0.0 exactly, so unmasked loads are safe.
        const int e0 = eb0 + n16;
        const int e1 = eb1 + n16;
#pragma unroll
        for (int v = 0; v < 8; ++v) {
            float h0 = hRow[v][e0];
            float h1 = hRow[v][e1];
            acc[v] += c0[v] * h0 + c1[v] * h1;
        }
    }

    // Reduce over the 16 lanes of each half (fixed xor tree -> deterministic)
#pragma unroll
    for (int v = 0; v < 8; ++v) {
        float x = acc[v];
        x += __shfl_xor(x, 1, 32);
        x += __shfl_xor(x, 2, 32);
        x += __shfl_xor(x, 4, 32);
        x += __shfl_xor(x, 8, 32);
        if (n16 == 0) sPart[wave][v + 8 * half] = x;
    }
    __syncthreads();

    if (tid < 16) {
        int s = sIdx[tid];
        if (s >= 0) {
            float sum = sBias;
#pragma unroll
            for (int w = 0; w < 8; ++w) sum += sPart[w][tid];
            out[s] = sum;
        }
    }
}

extern "C" void kernel_launch(void* const* d_in, const int* in_sizes, int n_in,
                              void* d_out, int out_size, void* d_ws, size_t ws_size,
                              hipStream_t stream) {
    (void)in_sizes; (void)n_in; (void)out_size; (void)ws_size;

    const float* head   = (const float*)d_in[0];
    const float* tail   = (const float*)d_in[1];
    const int*   rel_id = (const int*)d_in[2];
    const float* W      = (const float*)d_in[3];
    const float* b      = (const float*)d_in[4];
    float* out = (float*)d_out;

    int* counts = (int*)d_ws;        // R ints
    int* idx    = counts + R_N;      // R * B ints (~256 KB total workspace)

    zero_counts_kernel<<<1, 32, 0, stream>>>(counts);
    scatter_kernel<<<B_N / 256, 256, 0, stream>>>(rel_id, counts, idx);
    score_kernel<<<dim3(R_N, B_N / 16), 256, 0, stream>>>(
        head, tail, W, b, counts, idx, out);
}